// DiskLoss_32212254720671
// MI455X (gfx1250) — compile-verified
//
#include <hip/hip_runtime.h>
#include <hip/hip_bf16.h>

// Problem constants (B=2, C=33, H=W=64, K=64, SUB=10)
#define NSAMP 128          // B*K
#define GRID  640          // SUB*W == SUB*H
#define OFFC  160.0f       // SUB*W/4
#define NVTX  16           // (C-1)/2
#define NDISK 15           // len(range(0, C-3, 2))
#define PSTRIDE 80         // floats per per-sample param block
#define BPS   20           // row-blocks per sample (32 rows each)

typedef float v2f __attribute__((ext_vector_type(2)));
typedef float v8f __attribute__((ext_vector_type(8)));

// ---------------------------------------------------------------- init
__global__ void disk_init(unsigned int* acc) {
    int i = threadIdx.x;
    if (i < NSAMP * 3) acc[i] = 0u;
}

// ---------------------------------------------------------------- setup
// One thread per sample: gather p (channel-strided), read t, precompute
// polygon vertices + edge slopes + disk centers + r^2.
__global__ void disk_setup(const float* __restrict__ output,
                           const int* __restrict__ ind,
                           const float* __restrict__ target,
                           float* __restrict__ params) {
    int n = threadIdx.x;
    if (n >= NSAMP) return;
    int b  = n >> 6;                          // K = 64
    int hw = ind[n];
    const float* po = output + (size_t)b * 33 * 4096 + hw;   // p[c] = po[c*4096]
    const float* t  = target + n * 33;

    float x1[NVTX], y1[NVTX];
#pragma unroll
    for (int i = 0; i < NVTX; ++i) {
        x1[i] = 10.0f * t[2 * i]     + OFFC;
        y1[i] = 10.0f * t[2 * i + 1] + OFFC;
    }
    float* pb = params + n * PSTRIDE;
#pragma unroll
    for (int i = 0; i < NVTX; ++i) {
        int ip = (i + 1) & 15;
        pb[i]          = x1[i];
        pb[16 + i]     = y1[i];
        pb[32 + i]     = (x1[ip] - x1[i]) / (y1[ip] - y1[i] + 1e-9f);
    }
#pragma unroll
    for (int j = 0; j < NDISK; ++j) {
        pb[48 + j] = po[(2 * j)     * 4096] + OFFC;   // cx
        pb[63 + j] = po[(2 * j + 1) * 4096] + OFFC;   // cy
    }
    float r = ceilf(fabsf(po[32 * 4096]));
    pb[78] = r * r;
}

// ---------------------------------------------------------------- raster
// One workgroup = one sample x 32-row slab. 8 waves x 32 lanes; wave w owns
// rows rowBase + w + 8*rr. Per-row edge/disk state is wave-uniform.
__global__ void __launch_bounds__(256)
disk_raster(const float* __restrict__ params, unsigned int* __restrict__ acc) {
    __shared__ float sp[PSTRIDE];
    __shared__ unsigned int scnt[3];

    int bid     = blockIdx.x;
    int s       = bid / BPS;
    int rowBase = (bid % BPS) * 32;
    int tid     = threadIdx.x;

    if (tid < PSTRIDE) sp[tid] = params[s * PSTRIDE + tid];
    if (tid < 3)       scnt[tid] = 0u;
    __syncthreads();

    int w = tid >> 5, lane = tid & 31;

    float cxr[NDISK];
#pragma unroll
    for (int j = 0; j < NDISK; ++j) cxr[j] = sp[48 + j];
    float r2 = sp[78];

    unsigned int cg = 0, cd = 0, ci = 0;

    for (int rr = 0; rr < 4; ++rr) {
        int   row = rowBase + w + (rr << 3);
        float py  = (float)row;

        // Per-row edge precompute: fold the crossing flag into xint.
        float xe[NVTX];
#pragma unroll
        for (int k = 0; k < NVTX; ++k) {
            float y1 = sp[16 + k];
            float y2 = sp[16 + ((k + 1) & 15)];
            bool  cond = (y1 > py) != (y2 > py);
            float xint = sp[k] + (py - y1) * sp[32 + k];
            xe[k] = cond ? xint : -3.0e38f;   // never satisfies px < xe
        }
        float dy2[NDISK];
#pragma unroll
        for (int j = 0; j < NDISK; ++j) {
            float dy = py - sp[63 + j];
            dy2[j] = dy * dy;
        }

        for (int i = 0; i < GRID / 32; ++i) {
            float px = (float)(lane + (i << 5));
            unsigned int c = 0;
#pragma unroll
            for (int k = 0; k < NVTX; ++k) c += (px < xe[k]) ? 1u : 0u;
            unsigned int inside = c & 1u;

            unsigned int dk = 0u;
#pragma unroll
            for (int j = 0; j < NDISK; ++j) {
                float dx = px - cxr[j];
                dk |= (__builtin_fmaf(dx, dx, dy2[j]) <= r2) ? 1u : 0u;
            }
            cg += inside;
            cd += dk;
            ci += inside & dk;
        }
    }

    atomicAdd(&scnt[0], cg);
    atomicAdd(&scnt[1], cd);
    atomicAdd(&scnt[2], ci);
    __syncthreads();
    if (tid == 0) {
        atomicAdd(&acc[s * 3 + 0], scnt[0]);
        atomicAdd(&acc[s * 3 + 1], scnt[1]);
        atomicAdd(&acc[s * 3 + 2], scnt[2]);
    }
}

// ---------------------------------------------------------------- final
// One wave. WMMA f32 16x16x4 with B = ones reduces 64 per-lane values per
// instruction (row-sums independent of B layout); chained C accumulates the
// second 64. Documented C/D layout -> 7 adds + shfl_xor(16) completes.
__global__ void disk_final(const unsigned int* __restrict__ acc,
                           const int* __restrict__ mask,
                           float* __restrict__ out) {
    int L = threadIdx.x;   // 32 threads, full wave, EXEC all-ones

    float tv[4], mv[4];
#pragma unroll
    for (int q = 0; q < 4; ++q) {
        int   n   = L + 32 * q;
        float gt  = (float)acc[n * 3 + 0];
        float dk  = (float)acc[n * 3 + 1];
        float in  = (float)acc[n * 3 + 2];
        float uni = dk + gt - in;
        float iou = in / (uni + 1e-6f);
        mv[q] = (float)mask[n];
        tv[q] = iou * mv[q];
    }

    v2f ones; ones.x = 1.0f; ones.y = 1.0f;
    v8f aT = {0.f, 0.f, 0.f, 0.f, 0.f, 0.f, 0.f, 0.f};
    v8f aM = {0.f, 0.f, 0.f, 0.f, 0.f, 0.f, 0.f, 0.f};
    v2f a;

    a.x = tv[0]; a.y = tv[1];
    aT = __builtin_amdgcn_wmma_f32_16x16x4_f32(false, a, false, ones, (short)0, aT, false, false);
    a.x = tv[2]; a.y = tv[3];
    aT = __builtin_amdgcn_wmma_f32_16x16x4_f32(false, a, false, ones, (short)0, aT, false, false);
    a.x = mv[0]; a.y = mv[1];
    aM = __builtin_amdgcn_wmma_f32_16x16x4_f32(false, a, false, ones, (short)0, aM, false, false);
    a.x = mv[2]; a.y = mv[3];
    aM = __builtin_amdgcn_wmma_f32_16x16x4_f32(false, a, false, ones, (short)0, aM, false, false);

    float sT = aT[0] + aT[1] + aT[2] + aT[3] + aT[4] + aT[5] + aT[6] + aT[7];
    float sM = aM[0] + aM[1] + aM[2] + aM[3] + aM[4] + aM[5] + aM[6] + aM[7];
    float tot  = sT + __shfl_xor(sT, 16, 32);
    float totm = sM + __shfl_xor(sM, 16, 32);

    if (L == 0) out[0] = tot / (totm + 1e-6f);
}

// ---------------------------------------------------------------- launch
extern "C" void kernel_launch(void* const* d_in, const int* in_sizes, int n_in,
                              void* d_out, int out_size, void* d_ws, size_t ws_size,
                              hipStream_t stream) {
    const float* output = (const float*)d_in[0];   // (2,33,64,64) f32
    const int*   mask   = (const int*)  d_in[1];   // (2,64) i32
    const int*   ind    = (const int*)  d_in[2];   // (2,64) i32
    const float* target = (const float*)d_in[3];   // (2,64,33) f32
    // d_in[4] freq_mask unused by the reference loss.

    unsigned int* acc    = (unsigned int*)d_ws;                    // 128*3 u32
    float*        params = (float*)((char*)d_ws + 2048);           // 128*80 f32
    float*        out    = (float*)d_out;

    disk_init  <<<1,              NSAMP * 3, 0, stream>>>(acc);
    disk_setup <<<1,              NSAMP,     0, stream>>>(output, ind, target, params);
    disk_raster<<<NSAMP * BPS,    256,       0, stream>>>(params, acc);
    disk_final <<<1,              32,        0, stream>>>(acc, mask, out);
}